// _YatPerformerPolyBase_19232863552144
// MI455X (gfx1250) — compile-verified
//
#include <hip/hip_runtime.h>
#include <hip/hip_bf16.h>
#include <math.h>

// ---------------- problem constants (from reference) ----------------
constexpr int B_ = 2, T_ = 4096, E_ = 1024;
constexpr int H_ = 16, D_ = 64;
constexpr int M_ = 8, P_ = 64;            // R = 1
constexpr int CHUNK_ = 128;
constexpr int NCHUNK_ = T_ / CHUNK_;      // 32
constexpr int F_ = P_ * M_;               // 512
#define DELTA_ 1e-6f

typedef __attribute__((ext_vector_type(2))) float v2f;
typedef __attribute__((ext_vector_type(8))) float v8f;
typedef __attribute__((ext_vector_type(4))) unsigned int u32x4;
typedef __attribute__((ext_vector_type(8))) int i32x8;
typedef __attribute__((ext_vector_type(4))) int i32x4;

__device__ __forceinline__ v8f v8f_zero() {
  v8f z;
#pragma unroll
  for (int i = 0; i < 8; ++i) z[i] = 0.f;
  return z;
}

// D = A(16x4, f32) * B(4x16, f32) + C  (wave32, fp32 matrix pipe)
__device__ __forceinline__ v8f wmma4(v2f a, v2f b, v8f c) {
  return __builtin_amdgcn_wmma_f32_16x16x4_f32(false, a, false, b, (short)0, c,
                                               false, false);
}

// ---------------- Tensor Data Mover helpers (gfx1250) ----------------
// Low 32 bits of a generic shared pointer == LDS byte offset (aperture rule).
__device__ __forceinline__ unsigned lds_addr_of(const void* p) {
  return (unsigned)(unsigned long long)p;
}

// 2D tile DMA: global (row-major, rowStride elements of 4B) -> LDS.
// Optional LDS row padding via D# pad fields (interval/amount codes).
// tileCols in DWORDs per row, tileRows rows. pad_code<0 => no padding.
__device__ __forceinline__ void tdm_load_2d(const float* gtile, unsigned ldsByte,
                                            unsigned tileCols, unsigned tileRows,
                                            unsigned rowStride, int pad_interval_code) {
  const unsigned long long ga = (unsigned long long)gtile;
  u32x4 g0;
  g0[0] = 1u;                                          // count=1, user mode
  g0[1] = ldsByte;                                     // lds_addr (bytes)
  g0[2] = (unsigned)(ga & 0xffffffffull);              // global_addr[31:0]
  g0[3] = (unsigned)((ga >> 32) & 0x01ffffffull) | (2u << 30);  // [56:32]|type=2
  const unsigned dim0 = rowStride;                     // line length (elems)
  const unsigned dim1 = 0x7fffffffu;                   // plenty of rows (OOB off)
  i32x8 g1;
  unsigned flags = (2u << 16);                         // data_size = 4 bytes
  if (pad_interval_code >= 0)
    flags |= (1u << 20) | ((unsigned)pad_interval_code << 22);  // pad_amount=0 (1 DW)
  g1[0] = (int)flags;
  g1[1] = (int)((dim0 & 0xffffu) << 16);               // tensor_dim0[15:0]
  g1[2] = (int)((dim0 >> 16) | ((dim1 & 0xffffu) << 16));
  g1[3] = (int)((dim1 >> 16) | (tileCols << 16));      // tile_dim0
  g1[4] = (int)(tileRows & 0xffffu);                   // tile_dim1 (tile_dim2=0)
  g1[5] = (int)rowStride;                              // tensor_dim0_stride[31:0]
  g1[6] = 0;                                           // stride[47:32] | dim1_stride lo
  g1[7] = 0;
  i32x4 z4;
#pragma unroll
  for (int i = 0; i < 4; ++i) z4[i] = 0;
#if __clang_major__ >= 23
  i32x8 z8;
#pragma unroll
  for (int i = 0; i < 8; ++i) z8[i] = 0;
  __builtin_amdgcn_tensor_load_to_lds(g0, g1, z4, z4, z8, 0);
#else
  __builtin_amdgcn_tensor_load_to_lds(g0, g1, z4, z4, 0);
#endif
}

// =====================================================================
// Kernel 1/3: C(MxN) = A(MxK) * Bw(NxK)^T + bias   (both K-major, "NT")
// 128x128 block tile, 256 threads = 8 waves, each wave 32x64 (2x4 WMMA tiles).
// A/B K-tiles staged by the Tensor Data Mover, double-buffered on TENSORcnt.
// =====================================================================
constexpr int GT_M = 128, GT_N = 128, GT_K = 16, GT_KP = 17;

__global__ __launch_bounds__(256)
void wmma_gemm_nt_bias(const float* __restrict__ A, const float* __restrict__ Bw,
                       const float* __restrict__ bias, float* __restrict__ C,
                       int M, int N, int K) {
  __shared__ float sA[2][GT_M * GT_KP];   // TDM pad: 16 DW data + 1 DW pad
  __shared__ float sB[2][GT_N * GT_KP];

  const int tid  = threadIdx.x;
  const int wave = tid >> 5, lane = tid & 31;
  const int lr = lane & 15, lhi = lane >> 4;
  const int bm = blockIdx.y * GT_M, bn = blockIdx.x * GT_N;
  const int wr = (wave >> 1) * 32;   // 4 row-groups
  const int wc = (wave & 1) * 64;    // 2 col-groups

  v8f acc[2][4];
#pragma unroll
  for (int i = 0; i < 2; ++i)
#pragma unroll
    for (int j = 0; j < 4; ++j) acc[i][j] = v8f_zero();

  const int ntiles = K / GT_K;
  // prologue: DMA tile 0 into buffer 0 (pad_interval code 3 => every 16 DWORDs)
  if (wave == 0) {
    tdm_load_2d(A + (size_t)bm * K, lds_addr_of(&sA[0][0]), GT_K, GT_M, K, 3);
    tdm_load_2d(Bw + (size_t)bn * K, lds_addr_of(&sB[0][0]), GT_K, GT_N, K, 3);
  }

  for (int kt = 0; kt < ntiles; ++kt) {
    const int cur = kt & 1;
    if (wave == 0) {
      if (kt + 1 < ntiles) {  // DMA next tile, then gate on the older pair
        const int nxt = cur ^ 1;
        const int k1 = (kt + 1) * GT_K;
        tdm_load_2d(A + (size_t)bm * K + k1, lds_addr_of(&sA[nxt][0]), GT_K, GT_M, K, 3);
        tdm_load_2d(Bw + (size_t)bn * K + k1, lds_addr_of(&sB[nxt][0]), GT_K, GT_N, K, 3);
        __builtin_amdgcn_s_wait_tensorcnt((short)2);  // in-order: cur pair done
      } else {
        __builtin_amdgcn_s_wait_tensorcnt((short)0);
      }
    }
    __syncthreads();

    const float* __restrict__ tA = &sA[cur][0];
    const float* __restrict__ tB = &sB[cur][0];
#pragma unroll
    for (int kk = 0; kk < GT_K; kk += 4) {
      const int kb = kk + (lhi << 1);
      v2f af[2], bf[4];
#pragma unroll
      for (int i = 0; i < 2; ++i) {
        const int r = wr + i * 16 + lr;
        af[i].x = tA[r * GT_KP + kb];
        af[i].y = tA[r * GT_KP + kb + 1];
      }
#pragma unroll
      for (int j = 0; j < 4; ++j) {
        const int cidx = wc + j * 16 + lr;
        bf[j].x = tB[cidx * GT_KP + kb];
        bf[j].y = tB[cidx * GT_KP + kb + 1];
      }
#pragma unroll
      for (int i = 0; i < 2; ++i)
#pragma unroll
        for (int j = 0; j < 4; ++j) acc[i][j] = wmma4(af[i], bf[j], acc[i][j]);
    }
    __syncthreads();
  }

  // epilogue: C layout — VGPR e: rows (e, e+8) across lane halves, col = lane&15
#pragma unroll
  for (int i = 0; i < 2; ++i)
#pragma unroll
    for (int j = 0; j < 4; ++j) {
      const int gcol = bn + wc + j * 16 + lr;
      const float bv = bias[gcol];
#pragma unroll
      for (int e = 0; e < 8; ++e) {
        const int grow = bm + wr + i * 16 + e + (lhi << 3);
        C[(size_t)grow * N + gcol] = acc[i][j][e] + bv;
      }
    }
}

// =====================================================================
// Kernel 2: fused heads + l2norm + features + chunked causal scan
// 1 workgroup per (b,h): 1024 threads = 32 waves; kv_state lives in LDS.
// v-chunk arrives by TDM, overlapped with the q/k feature build.
// =====================================================================
constexpr int OFF_KV  = 0;                       // [512][64]
constexpr int OFF_KST = OFF_KV  + F_ * D_;       // [512]
constexpr int OFF_PW  = OFF_KST + F_;            // [64][64]  (d-major, p contig)
constexpr int OFF_OM  = OFF_PW  + D_ * P_;       // [64][8]
constexpr int OFF_XN  = OFF_OM  + D_ * M_;       // [128][65]
constexpr int OFF_PQ  = OFF_XN  + 128 * 65;      // [128][65] poly_q
constexpr int OFF_PK  = OFF_PQ  + 128 * 65;      // [128][65] poly_k
constexpr int OFF_FQ  = OFF_PK  + 128 * 65;      // [128][9]  prf_q
constexpr int OFF_FK  = OFF_FQ  + 128 * 9;       // [128][9]  prf_k
constexpr int OFF_V   = OFF_FK  + 128 * 9;       // [128][64]
constexpr int OFF_SC  = OFF_V   + 128 * 64;      // [128][65] score half-block
constexpr int OFF_NRM = OFF_SC  + 128 * 65;      // [128]
constexpr int ARENA_FLOATS = OFF_NRM + 128;      // 81792 floats = 327,168 B

__device__ __forceinline__ void build_features(
    const float* __restrict__ qkv, int b, int h, int t0, int which,
    float* __restrict__ s_xn, const float* __restrict__ s_pw,
    const float* __restrict__ s_om, float* __restrict__ s_poly,
    float* __restrict__ s_prf,
    float sq2s, float sNode, float prfScale, float polyScale,
    int tid, int wave, int lr, int lhi) {
  {  // load + l2norm: 8 lanes per row, float4-vectorized
    const int t = tid >> 3, part = tid & 7;
    const size_t rowbase =
        ((size_t)(b * T_ + t0 + t)) * (3 * E_) + (size_t)which * E_ + h * D_;
    const float4 v0 = *(const float4*)(qkv + rowbase + part * 8);
    const float4 v1 = *(const float4*)(qkv + rowbase + part * 8 + 4);
    float vals[8] = {v0.x, v0.y, v0.z, v0.w, v1.x, v1.y, v1.z, v1.w};
    float ss = 0.f;
#pragma unroll
    for (int j = 0; j < 8; ++j) ss += vals[j] * vals[j];
#pragma unroll
    for (int o = 4; o >= 1; o >>= 1) ss += __shfl_xor(ss, o, 8);
    const float inv = 1.f / fmaxf(sqrtf(ss), 1e-12f);
#pragma unroll
    for (int j = 0; j < 8; ++j) s_xn[t * 65 + part * 8 + j] = vals[j] * inv;
    if (t0 + CHUNK_ < T_) {  // gfx1250 global_prefetch_b8: next chunk's rows
      __builtin_prefetch(qkv + rowbase + (size_t)CHUNK_ * 3 * E_, 0, 1);
    }
  }
  __syncthreads();
  {  // poly = (xn @ poly_w)^2 / sqrt(P) : 8x4 WMMA tile grid, 1 tile / wave
    const int ti = wave >> 2, pi = wave & 3;
    const int r = ti * 16 + lr, col = pi * 16 + lr;
    v8f acc = v8f_zero();
#pragma unroll
    for (int k0 = 0; k0 < D_; k0 += 4) {
      const int kb = k0 + (lhi << 1);
      v2f a, bb;
      a.x = s_xn[r * 65 + kb];       a.y = s_xn[r * 65 + kb + 1];
      bb.x = s_pw[kb * P_ + col];    bb.y = s_pw[(kb + 1) * P_ + col];
      acc = wmma4(a, bb, acc);
    }
#pragma unroll
    for (int e = 0; e < 8; ++e) {
      const int rr = ti * 16 + e + (lhi << 3);
      const float pv = acc[e];
      s_poly[rr * 65 + col] = pv * pv * polyScale;
    }
  }
  {  // prf = exp(clip(proj*sqrt(2s)-s)) * sqrt(w)/sqrt(M): (t,m) per thread
    const int t = tid >> 3, m = tid & 7;
    float pr = 0.f;
#pragma unroll 8
    for (int d = 0; d < D_; ++d) pr += s_xn[t * 65 + d] * s_om[d * M_ + m];
    const float z = fminf(fmaxf(pr * sq2s - sNode, -20.f), 20.f);
    s_prf[t * 9 + m] = __expf(z) * prfScale;
  }
  __syncthreads();
}

__global__ __launch_bounds__(1024, 1)
void yat_attn(const float* __restrict__ qkv, const float* __restrict__ omega,
              const float* __restrict__ poly_w, const float* __restrict__ qnodes,
              const float* __restrict__ qweights, float* __restrict__ o_out) {
  extern __shared__ float sm[];
  float* s_kv  = sm + OFF_KV;
  float* s_kst = sm + OFF_KST;
  float* s_pw  = sm + OFF_PW;
  float* s_om  = sm + OFF_OM;
  float* s_xn  = sm + OFF_XN;
  float* s_pq  = sm + OFF_PQ;
  float* s_pk  = sm + OFF_PK;
  float* s_fq  = sm + OFF_FQ;
  float* s_fk  = sm + OFF_FK;
  float* s_v   = sm + OFF_V;
  float* s_sc  = sm + OFF_SC;
  float* s_nrm = sm + OFF_NRM;

  const int h = blockIdx.x, b = blockIdx.y;
  const int tid = threadIdx.x;
  const int wave = tid >> 5, lane = tid & 31;
  const int lr = lane & 15, lhi = lane >> 4;

  const float sNode = qnodes[0];
  const float sq2s = sqrtf(fmaxf(2.f * sNode, 0.f));
  const float prfScale = sqrtf(fmaxf(qweights[0], 0.f)) * 0.3535533905932738f;
  const float polyScale = 0.125f;  // 1/sqrt(P)

  for (int i = tid; i < D_ * P_; i += 1024) {
    const int d = i >> 6, p = i & 63;
    s_pw[i] = poly_w[((size_t)h * D_ + d) * P_ + p];
  }
  for (int i = tid; i < D_ * M_; i += 1024) {
    const int d = i >> 3, m = i & 7;
    s_om[i] = omega[((size_t)h * D_ + d) * M_ + m];  // R = 1
  }
  for (int i = tid; i < F_ * D_; i += 1024) s_kv[i] = 0.f;
  if (tid < F_) s_kst[tid] = 0.f;
  __syncthreads();

  for (int c = 0; c < NCHUNK_; ++c) {
    const int t0 = c * CHUNK_;

    // kick off the v-chunk DMA (128 x 64 f32, row stride 3E); it overlaps the
    // whole q/k feature build below. End-of-chunk barrier guarantees nobody is
    // still reading s_v from the previous chunk when this lands.
    if (wave == 0) {
      const float* vtile =
          qkv + ((size_t)(b * T_ + t0)) * (3 * E_) + 2 * (size_t)E_ + h * D_;
      tdm_load_2d(vtile, lds_addr_of(s_v), D_, CHUNK_, 3 * E_, -1 /*no pad*/);
    }

    build_features(qkv, b, h, t0, 0, s_xn, s_pw, s_om, s_pq, s_fq,
                   sq2s, sNode, prfScale, polyScale, tid, wave, lr, lhi);
    build_features(qkv, b, h, t0, 1, s_xn, s_pw, s_om, s_pk, s_fk,
                   sq2s, sNode, prfScale, polyScale, tid, wave, lr, lhi);

    {  // nrm[t] = DELTA + qf[t,:] . k_state   (qf built on the fly)
      const int t = tid >> 3, part = tid & 7;
      const int f0 = part * 64;
      float sacc = 0.f;
#pragma unroll 8
      for (int ff = 0; ff < 64; ++ff) {
        const int f = f0 + ff;
        sacc += s_pq[t * 65 + (f >> 3)] * s_fq[t * 9 + (f & 7)] * s_kst[f];
      }
#pragma unroll
      for (int o = 4; o >= 1; o >>= 1) sacc += __shfl_xor(sacc, o, 8);
      if (part == 0) s_nrm[t] = DELTA_ + sacc;
    }
    if (wave == 0) __builtin_amdgcn_s_wait_tensorcnt((short)0);  // s_v ready
    __syncthreads();

    // ---- context accumulation: wave owns (t-tile, d-tile) of 8x4 grid ----
    const int ti = wave >> 2, di = wave & 3;
    const int r_t = ti * 16 + lr, col_d = di * 16 + lr;
    v8f acc = v8f_zero();

    // ctx_hist = qf @ kv_state   (K = F = 512, A built on the fly)
#pragma unroll 4
    for (int f0 = 0; f0 < F_; f0 += 4) {
      const int kb = f0 + (lhi << 1);
      v2f a, bb;
      a.x = s_pq[r_t * 65 + (kb >> 3)] * s_fq[r_t * 9 + (kb & 7)];
      a.y = s_pq[r_t * 65 + ((kb + 1) >> 3)] * s_fq[r_t * 9 + ((kb + 1) & 7)];
      bb.x = s_kv[kb * D_ + col_d];
      bb.y = s_kv[(kb + 1) * D_ + col_d];
      acc = wmma4(a, bb, acc);
    }

    // intra-chunk causal part in two 64-col halves (scores factorize!)
#pragma unroll 1
    for (int half = 0; half < 2; ++half) {
      {  // scores tile (sti, ssi) per wave: Spoly ⊙ Sprf, masked
        const int sti = wave >> 2, ssi = wave & 3;
        const int scol_l = ssi * 16 + lr;            // 0..63 within half
        const int s_ch = half * 64 + scol_l;         // 0..127 chunk-local
        v8f aS = v8f_zero(), aP = v8f_zero();
        if (sti * 16 + 15 >= half * 64 + ssi * 16) {  // wave-uniform guard
          const int r = sti * 16 + lr;
#pragma unroll
          for (int k0 = 0; k0 < P_; k0 += 4) {
            const int kb = k0 + (lhi << 1);
            v2f a, bb;
            a.x = s_pq[r * 65 + kb];       a.y = s_pq[r * 65 + kb + 1];
            bb.x = s_pk[s_ch * 65 + kb];   bb.y = s_pk[s_ch * 65 + kb + 1];
            aS = wmma4(a, bb, aS);
          }
#pragma unroll
          for (int k0 = 0; k0 < M_; k0 += 4) {
            const int kb = k0 + (lhi << 1);
            v2f a, bb;
            a.x = s_fq[r * 9 + kb];        a.y = s_fq[r * 9 + kb + 1];
            bb.x = s_fk[s_ch * 9 + kb];    bb.y = s_fk[s_ch * 9 + kb + 1];
            aP = wmma4(a, bb, aP);
          }
        }
#pragma unroll
        for (int e = 0; e < 8; ++e) {
          const int rr = sti * 16 + e + (lhi << 3);
          const float val = (rr >= s_ch) ? aS[e] * aP[e] : 0.f;
          s_sc[rr * 65 + scol_l] = val;
        }
      }
      __syncthreads();
      {  // row-sum of this score half into nrm
        const int t = tid >> 3, part = tid & 7;
        float sacc = 0.f;
#pragma unroll
        for (int j = 0; j < 8; ++j) sacc += s_sc[t * 65 + part * 8 + j];
#pragma unroll
        for (int o = 4; o >= 1; o >>= 1) sacc += __shfl_xor(sacc, o, 8);
        if (part == 0) s_nrm[t] += sacc;
      }
      {  // ctx_intra partial: scores(half) @ v(half)  (K = 64)
#pragma unroll
        for (int k0 = 0; k0 < 64; k0 += 4) {
          const int kb = k0 + (lhi << 1);
          v2f a, bb;
          a.x = s_sc[r_t * 65 + kb];  a.y = s_sc[r_t * 65 + kb + 1];
          bb.x = s_v[(half * 64 + kb) * 64 + col_d];
          bb.y = s_v[(half * 64 + kb + 1) * 64 + col_d];
          acc = wmma4(a, bb, acc);
        }
      }
      __syncthreads();
    }

    {  // o = ctx / nrm  → (B,T,H,D) == (B,T,E) layout for the output GEMM
#pragma unroll
      for (int e = 0; e < 8; ++e) {
        const int rr = ti * 16 + e + (lhi << 3);
        const float ov = acc[e] / s_nrm[rr];
        o_out[((size_t)(b * T_ + t0 + rr)) * E_ + h * D_ + col_d] = ov;
      }
    }
    __syncthreads();

    {  // kv_state += kf^T @ v : wave = f-tile (32), loop 4 d-tiles; C from LDS
      const int ftile = wave;
      const int fr = ftile * 16 + lr;  // f row of A fragment
#pragma unroll 1
      for (int dt = 0; dt < 4; ++dt) {
        const int col = dt * 16 + lr;
        v8f ckv;
#pragma unroll
        for (int e = 0; e < 8; ++e)
          ckv[e] = s_kv[(ftile * 16 + e + (lhi << 3)) * D_ + col];
#pragma unroll 2
        for (int s0 = 0; s0 < CHUNK_; s0 += 4) {
          const int kb = s0 + (lhi << 1);
          v2f a, bb;
          a.x = s_pk[kb * 65 + (fr >> 3)] * s_fk[kb * 9 + (fr & 7)];
          a.y = s_pk[(kb + 1) * 65 + (fr >> 3)] * s_fk[(kb + 1) * 9 + (fr & 7)];
          bb.x = s_v[kb * 64 + col];
          bb.y = s_v[(kb + 1) * 64 + col];
          ckv = wmma4(a, bb, ckv);
        }
#pragma unroll
        for (int e = 0; e < 8; ++e)
          s_kv[(ftile * 16 + e + (lhi << 3)) * D_ + col] = ckv[e];
      }
    }
    if (tid < F_) {  // k_state += sum_s kf[s,f]
      const int p = tid >> 3, m = tid & 7;
      float sacc = 0.f;
#pragma unroll 4
      for (int s = 0; s < CHUNK_; ++s) sacc += s_pk[s * 65 + p] * s_fk[s * 9 + m];
      s_kst[tid] += sacc;
    }
    __syncthreads();
  }
}

// =====================================================================
extern "C" void kernel_launch(void* const* d_in, const int* in_sizes, int n_in,
                              void* d_out, int out_size, void* d_ws,
                              size_t ws_size, hipStream_t stream) {
  (void)in_sizes; (void)n_in; (void)out_size; (void)ws_size;
  const float* x      = (const float*)d_in[0];
  const float* qkv_w  = (const float*)d_in[1];
  const float* qkv_b  = (const float*)d_in[2];
  const float* out_w  = (const float*)d_in[3];
  const float* out_b  = (const float*)d_in[4];
  const float* omega  = (const float*)d_in[5];
  const float* poly_w = (const float*)d_in[6];
  const float* qnodes = (const float*)d_in[7];
  const float* qwts   = (const float*)d_in[8];
  float* out = (float*)d_out;

  float* ws_qkv = (float*)d_ws;                          // (B*T, 3E) = 100.7 MB
  float* ws_o   = ws_qkv + (size_t)B_ * T_ * 3 * E_;     // (B*T, E)  =  33.6 MB

  // 1) qkv = x @ qkv_w^T + qkv_b
  wmma_gemm_nt_bias<<<dim3((3 * E_) / GT_N, (B_ * T_) / GT_M), dim3(256), 0,
                      stream>>>(x, qkv_w, qkv_b, ws_qkv, B_ * T_, 3 * E_, E_);

  // 2) fused features + chunked causal linear-attention scan (kv in LDS)
  const size_t lds = (size_t)ARENA_FLOATS * sizeof(float);  // 327,168 B
  yat_attn<<<dim3(H_, B_), dim3(1024), lds, stream>>>(ws_qkv, omega, poly_w,
                                                      qnodes, qwts, ws_o);

  // 3) out = o @ out_w^T + out_b
  wmma_gemm_nt_bias<<<dim3(E_ / GT_N, (B_ * T_) / GT_M), dim3(256), 0,
                      stream>>>(ws_o, out_w, out_b, out, B_ * T_, E_, E_);
}